// GCconv_8993661518260
// MI455X (gfx1250) — compile-verified
//
#include <hip/hip_runtime.h>
#include <hip/hip_bf16.h>

typedef __attribute__((ext_vector_type(2))) float v2f;
typedef __attribute__((ext_vector_type(8))) float v8f;

#define NB   16
#define NC   512
#define NH   64
#define NW   64
#define NHW  4096
#define NKK  49
#define NKKP 64
#define LN_EPS 1e-5f

__device__ __forceinline__ v8f wmma_f32(v2f a, v2f b, v8f c) {
    return __builtin_amdgcn_wmma_f32_16x16x4_f32(false, a, false, b, (short)0, c,
                                                 false, false);
}

// ---------------------------------------------------------------------------
// Kernel 1: key logits  L[b, m, n] = sum_c Kw[m,c] * x[b,c,n] + Kb[m]
// Workgroup covers M=64 x N=128. 8 waves in a 2x4 grid; each wave computes a
// 2x2 block of 16x16 tiles (2 A-frags + 2 B-frags -> 4 WMMA per K-step).
// Pad rows (m >= 49) read a clamped Kw row: values are finite garbage that is
// never consumed (ln_relu zeroes pad columns) -- no divergence in the loop.
// ---------------------------------------------------------------------------
__global__ void __launch_bounds__(256) key_gemm(const float* __restrict__ x,
                                                const float* __restrict__ Kw,
                                                const float* __restrict__ Kb,
                                                float* __restrict__ S) {
    const int lane = threadIdx.x & 31;
    const int wave = threadIdx.x >> 5;
    const int half = lane >> 4;
    const int l16  = lane & 15;

    const int b  = blockIdx.y;
    const int mg = wave & 1;          // M block: rows [mg*32, mg*32+32)
    const int ng = wave >> 1;         // N block: cols [ng*32, ng*32+32)

    const int m0 = mg * 32 + l16;
    const int m1 = m0 + 16;
    const int m0c = (m0 < NKK) ? m0 : (NKK - 1);
    const int m1c = (m1 < NKK) ? m1 : (NKK - 1);
    const int n0 = blockIdx.x * 128 + ng * 32 + l16;
    const int n1 = n0 + 16;

    const float* xb  = x + (size_t)b * NC * NHW;
    const float* a0p = Kw + (size_t)m0c * NC;
    const float* a1p = Kw + (size_t)m1c * NC;

    v8f acc00 = {0.f,0.f,0.f,0.f,0.f,0.f,0.f,0.f};
    v8f acc01 = acc00, acc10 = acc00, acc11 = acc00;

#pragma unroll 4
    for (int k0 = 0; k0 < NC; k0 += 4) {
        const int ka = k0 + 2 * half;
        v2f a0 = *reinterpret_cast<const v2f*>(a0p + ka);
        v2f a1 = *reinterpret_cast<const v2f*>(a1p + ka);
        const float* xk = xb + (size_t)ka * NHW;
        v2f b0, b1;
        b0.x = xk[n0];       b0.y = xk[NHW + n0];
        b1.x = xk[n1];       b1.y = xk[NHW + n1];
        acc00 = wmma_f32(a0, b0, acc00);
        acc01 = wmma_f32(a0, b1, acc01);
        acc10 = wmma_f32(a1, b0, acc10);
        acc11 = wmma_f32(a1, b1, acc11);
    }

    float* Sb = S + (size_t)b * NKKP * NHW;
#pragma unroll
    for (int r = 0; r < 8; ++r) {
        const int ra = mg * 32 + r + half * 8;
        const int rb = ra + 16;
        const float biasa = (ra < NKK) ? Kb[ra] : 0.f;
        const float biasb = (rb < NKK) ? Kb[rb] : 0.f;
        Sb[(size_t)ra * NHW + n0] = acc00[r] + biasa;
        Sb[(size_t)ra * NHW + n1] = acc01[r] + biasa;
        Sb[(size_t)rb * NHW + n0] = acc10[r] + biasb;
        Sb[(size_t)rb * NHW + n1] = acc11[r] + biasb;
    }
}

// ---------------------------------------------------------------------------
// Kernel 2: softmax over the spatial axis (4096) for each of B*49 rows.
// ---------------------------------------------------------------------------
__global__ void __launch_bounds__(256) softmax_rows(float* __restrict__ S) {
    const int row = blockIdx.x;            // 0 .. B*NKK-1
    const int b   = row / NKK;
    const int kk  = row % NKK;
    float* p = S + (size_t)b * NKKP * NHW + (size_t)kk * NHW;

    __shared__ float red[256];
    const int tid = threadIdx.x;

    float m = -3.402823466e38f;
    for (int i = tid; i < NHW; i += 256) m = fmaxf(m, p[i]);
    red[tid] = m;
    __syncthreads();
    for (int s = 128; s > 0; s >>= 1) {
        if (tid < s) red[tid] = fmaxf(red[tid], red[tid + s]);
        __syncthreads();
    }
    m = red[0];
    __syncthreads();

    float acc = 0.f;
    for (int i = tid; i < NHW; i += 256) acc += __expf(p[i] - m);
    red[tid] = acc;
    __syncthreads();
    for (int s = 128; s > 0; s >>= 1) {
        if (tid < s) red[tid] += red[tid + s];
        __syncthreads();
    }
    const float inv = 1.f / red[0];

    for (int i = tid; i < NHW; i += 256) p[i] = __expf(p[i] - m) * inv;
}

// ---------------------------------------------------------------------------
// Kernel 3: F0[b, c, kk] = sum_n x[b,c,n] * S[b,kk,n]
// M = 512 (c), N = 64 (kk padded), K = 4096.
// 8 waves, each owns 1 M-tile and all 4 N-tiles (1 A load -> 4 WMMA).
// ---------------------------------------------------------------------------
__global__ void __launch_bounds__(256) filters_gemm(const float* __restrict__ x,
                                                    const float* __restrict__ S,
                                                    float* __restrict__ F0) {
    const int lane = threadIdx.x & 31;
    const int wave = threadIdx.x >> 5;
    const int half = lane >> 4;
    const int l16  = lane & 15;

    const int b = blockIdx.y;
    const int m = blockIdx.x * 128 + wave * 16 + l16;   // channel row

    const float* xb = x + (size_t)b * NC * NHW;
    const float* Sb = S + (size_t)b * NKKP * NHW;
    const float* arow = xb + (size_t)m * NHW;
    const float* br0 = Sb + (size_t)(l16)      * NHW;   // B[k][n] = S[n][k]
    const float* br1 = Sb + (size_t)(16 + l16) * NHW;
    const float* br2 = Sb + (size_t)(32 + l16) * NHW;
    const float* br3 = Sb + (size_t)(48 + l16) * NHW;

    v8f acc0 = {0.f,0.f,0.f,0.f,0.f,0.f,0.f,0.f};
    v8f acc1 = acc0, acc2 = acc0, acc3 = acc0;

#pragma unroll 4
    for (int k0 = 0; k0 < NHW; k0 += 4) {
        const int ka = k0 + 2 * half;
        v2f a  = *reinterpret_cast<const v2f*>(arow + ka);
        v2f b0 = *reinterpret_cast<const v2f*>(br0 + ka);
        v2f b1 = *reinterpret_cast<const v2f*>(br1 + ka);
        v2f b2 = *reinterpret_cast<const v2f*>(br2 + ka);
        v2f b3 = *reinterpret_cast<const v2f*>(br3 + ka);
        acc0 = wmma_f32(a, b0, acc0);
        acc1 = wmma_f32(a, b1, acc1);
        acc2 = wmma_f32(a, b2, acc2);
        acc3 = wmma_f32(a, b3, acc3);
    }

    float* Fb = F0 + (size_t)b * NC * NKKP;
#pragma unroll
    for (int r = 0; r < 8; ++r) {
        const int row = blockIdx.x * 128 + wave * 16 + r + half * 8;
        float* fr = Fb + (size_t)row * NKKP + l16;
        fr[0]  = acc0[r];
        fr[16] = acc1[r];
        fr[32] = acc2[r];
        fr[48] = acc3[r];
    }
}

// ---------------------------------------------------------------------------
// Kernel 4 (conv0 / conv1): Fout[b,o,kk] = sum_c W[o,c]*Fin[b,c,kk] + bias[o]
// M = 512, N = 64, K = 512.
// 8 waves, each owns 1 M-tile and all 4 N-tiles (1 A load -> 4 WMMA).
// ---------------------------------------------------------------------------
__global__ void __launch_bounds__(256) cconv_gemm(const float* __restrict__ W,
                                                  const float* __restrict__ bias,
                                                  const float* __restrict__ Fin,
                                                  float* __restrict__ Fout) {
    const int lane = threadIdx.x & 31;
    const int wave = threadIdx.x >> 5;
    const int half = lane >> 4;
    const int l16  = lane & 15;

    const int b = blockIdx.y;
    const int m = blockIdx.x * 128 + wave * 16 + l16;   // output channel

    const float* Fb = Fin + (size_t)b * NC * NKKP;
    const float* arow = W + (size_t)m * NC;

    v8f acc0 = {0.f,0.f,0.f,0.f,0.f,0.f,0.f,0.f};
    v8f acc1 = acc0, acc2 = acc0, acc3 = acc0;

#pragma unroll 4
    for (int k0 = 0; k0 < NC; k0 += 4) {
        const int ka = k0 + 2 * half;
        v2f a = *reinterpret_cast<const v2f*>(arow + ka);
        const float* Fk = Fb + (size_t)ka * NKKP + l16;
        v2f b0, b1, b2, b3;
        b0.x = Fk[0];          b0.y = Fk[NKKP];
        b1.x = Fk[16];         b1.y = Fk[NKKP + 16];
        b2.x = Fk[32];         b2.y = Fk[NKKP + 32];
        b3.x = Fk[48];         b3.y = Fk[NKKP + 48];
        acc0 = wmma_f32(a, b0, acc0);
        acc1 = wmma_f32(a, b1, acc1);
        acc2 = wmma_f32(a, b2, acc2);
        acc3 = wmma_f32(a, b3, acc3);
    }

    float* Fo = Fout + (size_t)b * NC * NKKP;
#pragma unroll
    for (int r = 0; r < 8; ++r) {
        const int row = blockIdx.x * 128 + wave * 16 + r + half * 8;
        const float bs = bias[row];
        float* fr = Fo + (size_t)row * NKKP + l16;
        fr[0]  = acc0[r] + bs;
        fr[16] = acc1[r] + bs;
        fr[32] = acc2[r] + bs;
        fr[48] = acc3[r] + bs;
    }
}

// ---------------------------------------------------------------------------
// Kernel 5: LayerNorm over [C, 7, 7] (only kk < 49 valid) + ReLU.
// One workgroup per batch. Zeroes padded kk columns 49..63.
// ---------------------------------------------------------------------------
__global__ void __launch_bounds__(256) ln_relu(const float* __restrict__ Fin,
                                               const float* __restrict__ lnw,
                                               const float* __restrict__ lnb,
                                               float* __restrict__ Fout) {
    const int b   = blockIdx.x;
    const int tid = threadIdx.x;
    const float* Fb = Fin + (size_t)b * NC * NKKP;
    float* Fo = Fout + (size_t)b * NC * NKKP;

    __shared__ float rs[256];
    __shared__ float rs2[256];

    const int NVALID = NC * NKK;   // 25088
    float s = 0.f, s2 = 0.f;
    for (int i = tid; i < NVALID; i += 256) {
        const int c  = i / NKK;
        const int kk = i - c * NKK;
        const float v = Fb[(size_t)c * NKKP + kk];
        s += v;
        s2 += v * v;
    }
    rs[tid] = s; rs2[tid] = s2;
    __syncthreads();
    for (int st = 128; st > 0; st >>= 1) {
        if (tid < st) { rs[tid] += rs[tid + st]; rs2[tid] += rs2[tid + st]; }
        __syncthreads();
    }
    const float mean = rs[0] / (float)NVALID;
    const float var  = rs2[0] / (float)NVALID - mean * mean;
    const float rstd = rsqrtf(var + LN_EPS);

    for (int j = tid; j < NC * NKKP; j += 256) {
        const int c  = j >> 6;
        const int kk = j & 63;
        float v = 0.f;
        if (kk < NKK) {
            v = (Fb[j] - mean) * rstd * lnw[(size_t)c * NKK + kk] +
                lnb[(size_t)c * NKK + kk];
            v = fmaxf(v, 0.f);
        }
        Fo[j] = v;
    }
}

// ---------------------------------------------------------------------------
// Kernel 6: per-(b,c) depthwise 7x7 conv, zero-padded by 3.
// One workgroup per feature map; map staged in LDS (16 KB of 320 KB WGP LDS).
// ---------------------------------------------------------------------------
__global__ void __launch_bounds__(256) depthwise7(const float* __restrict__ x,
                                                  const float* __restrict__ Filt,
                                                  float* __restrict__ out) {
    const int map = blockIdx.x;            // b*NC + c
    const int tid = threadIdx.x;

    __shared__ float tile[NHW];
    __shared__ float flt[NKK];

    const float* xin = x + (size_t)map * NHW;
    for (int i = tid; i < NHW; i += 256) tile[i] = xin[i];
    if (tid < NKK) flt[tid] = Filt[(size_t)map * NKKP + tid];
    __syncthreads();

    float* op = out + (size_t)map * NHW;
    for (int i = tid; i < NHW; i += 256) {
        const int y  = i >> 6;
        const int xx = i & 63;
        float acc = 0.f;
#pragma unroll
        for (int dy = 0; dy < 7; ++dy) {
            const int yy = y + dy - 3;
            if (yy < 0 || yy >= NH) continue;
#pragma unroll
            for (int dx = 0; dx < 7; ++dx) {
                const int xc = xx + dx - 3;
                if (xc < 0 || xc >= NW) continue;
                acc += tile[yy * NW + xc] * flt[dy * 7 + dx];
            }
        }
        op[i] = acc;
    }
}

// ---------------------------------------------------------------------------
extern "C" void kernel_launch(void* const* d_in, const int* in_sizes, int n_in,
                              void* d_out, int out_size, void* d_ws, size_t ws_size,
                              hipStream_t stream) {
    const float* x       = (const float*)d_in[0];
    const float* K_w     = (const float*)d_in[1];
    const float* K_b     = (const float*)d_in[2];
    const float* conv0_w = (const float*)d_in[3];
    const float* conv0_b = (const float*)d_in[4];
    const float* ln_w    = (const float*)d_in[5];
    const float* ln_b    = (const float*)d_in[6];
    const float* conv1_w = (const float*)d_in[7];
    const float* conv1_b = (const float*)d_in[8];
    float* out = (float*)d_out;

    float* S  = (float*)d_ws;                                  // 16 MiB
    float* fA = S + (size_t)NB * NKKP * NHW;                   // 2 MiB
    float* fB = fA + (size_t)NB * NC * NKKP;                   // 2 MiB

    // 1) key logits:  M=64 x N=4096 per batch, WG tile 64x128
    key_gemm<<<dim3(NHW / 128, NB), 256, 0, stream>>>(x, K_w, K_b, S);
    // 2) softmax over spatial
    softmax_rows<<<NB * NKK, 256, 0, stream>>>(S);
    // 3) attention-pooled filters: WG tile 128x64
    filters_gemm<<<dim3(NC / 128, NB), 256, 0, stream>>>(x, S, fA);
    // 4) conv0
    cconv_gemm<<<dim3(NC / 128, NB), 256, 0, stream>>>(conv0_w, conv0_b, fA, fB);
    // 5) LayerNorm + ReLU
    ln_relu<<<NB, 256, 0, stream>>>(fB, ln_w, ln_b, fA);
    // 6) conv1
    cconv_gemm<<<dim3(NC / 128, NB), 256, 0, stream>>>(conv1_w, conv1_b, fA, fB);
    // 7) depthwise 7x7 with dynamic per-map filters
    depthwise7<<<NB * NC, 256, 0, stream>>>(x, fB, out);
}